// ChebConv_39977555591181
// MI455X (gfx1250) — compile-verified
//
#include <hip/hip_runtime.h>

#define N_NODES 50000
#define N_EDGES 800000
#define D_FEAT  128
#define UNITS   128
#define N_HOPS  3

typedef __attribute__((ext_vector_type(2))) float v2f;
typedef __attribute__((ext_vector_type(8))) float v8f;

// ---------------- kernel 1: out[n,u] = sum_i bias[i][u] ----------------
__global__ void cheb_init_out(float* __restrict__ out, const float* __restrict__ bias) {
    int idx = blockIdx.x * blockDim.x + threadIdx.x;
    if (idx >= N_NODES * UNITS) return;
    int u = idx & (UNITS - 1);
    out[idx] = bias[u] + bias[UNITS + u] + bias[2 * UNITS + u];
}

// ---------------- kernel 2: h[i] = x @ W[i]  (FP32 WMMA 16x16x4) -------
// One 16x16 output tile per wave32. K loop: 32 steps of 4.
__global__ void cheb_gemm(const float* __restrict__ x,
                          const float* __restrict__ W,
                          float* __restrict__ h) {
    const int waveInBlock = threadIdx.x >> 5;
    const int lane        = threadIdx.x & 31;
    const int  mTiles      = N_NODES / 16;           // 3125
    const int  nTiles      = UNITS   / 16;           // 8
    const long tilesPerHop = (long)mTiles * nTiles;  // 25000
    const long tile        = (long)blockIdx.x * 8 + waveInBlock;
    if (tile >= tilesPerHop * N_HOPS) return;        // wave-uniform guard

    const int  hop = (int)(tile / tilesPerHop);
    const long t   = tile % tilesPerHop;
    const int  m0  = (int)(t / nTiles) * 16;
    const int  n0  = (int)(t % nTiles) * 16;

    const float* Wh = W + (size_t)hop * D_FEAT * UNITS;
    float*       hh = h + (size_t)hop * N_NODES * UNITS;

    // ISA A-frag layout (16x4 f32): lanes 0-15 hold K=0/1, lanes 16-31 hold K=2/3
    const int half = lane >> 4;         // 0 or 1
    const int lmod = lane & 15;
    const float* arow = x + (size_t)(m0 + lmod) * D_FEAT;

    v8f c = {};
    #pragma unroll 4
    for (int k = 0; k < D_FEAT; k += 4) {
        const int ka = k + 2 * half;
        v2f a, b;
        a.x = arow[ka];
        a.y = arow[ka + 1];
        b.x = Wh[(size_t)ka * UNITS + n0 + lmod];
        b.y = Wh[(size_t)(ka + 1) * UNITS + n0 + lmod];
        // (neg_a, A, neg_b, B, c_mod, C, reuse_a, reuse_b)
        c = __builtin_amdgcn_wmma_f32_16x16x4_f32(false, a, false, b,
                                                  (short)0, c, false, false);
    }

    // C layout: VGPR r -> row m0 + r + 8*half, col n0 + lmod
    #pragma unroll
    for (int r = 0; r < 8; ++r) {
        hh[(size_t)(m0 + r + 8 * half) * UNITS + n0 + lmod] = c[r];
    }
}

// ---------------- kernel 3: edge gather + scatter-add ------------------
// One wave per (hop, edge). Lane l handles features [4l, 4l+4):
// 512B coalesced gather from h[src], 4 hardware f32 atomics into out[dst].
__global__ void cheb_scatter(const float* __restrict__ h,
                             const float* __restrict__ ew,
                             const long long* __restrict__ eidx,
                             float* __restrict__ out) {
    const int lane = threadIdx.x & 31;
    const long long wid = ((long long)blockIdx.x * blockDim.x + threadIdx.x) >> 5;
    const long long total = (long long)N_HOPS * N_EDGES;
    if (wid >= total) return;                        // wave-uniform guard

    const int       hop = (int)(wid / N_EDGES);
    const long long e   = wid % N_EDGES;

    const long long* base = eidx + (long long)hop * 2 * N_EDGES;
    const long long src = base[e];
    const long long dst = base[N_EDGES + e];
    const float     w   = ew[(long long)hop * N_EDGES + e];

    const float4* hv = (const float4*)(h + (size_t)hop * N_NODES * UNITS
                                         + (size_t)src * UNITS);
    const float4 v = hv[lane];

    float* o = out + (size_t)dst * UNITS + (size_t)lane * 4;
    unsafeAtomicAdd(o + 0, v.x * w);   // global_atomic_add_f32
    unsafeAtomicAdd(o + 1, v.y * w);
    unsafeAtomicAdd(o + 2, v.z * w);
    unsafeAtomicAdd(o + 3, v.w * w);
}

// ---------------- kernel 4: in-place ReLU ------------------------------
__global__ void cheb_relu(float* __restrict__ out) {
    int idx = blockIdx.x * blockDim.x + threadIdx.x;
    if (idx < N_NODES * UNITS) out[idx] = fmaxf(out[idx], 0.0f);
}

extern "C" void kernel_launch(void* const* d_in, const int* in_sizes, int n_in,
                              void* d_out, int out_size, void* d_ws, size_t ws_size,
                              hipStream_t stream) {
    const float*     x      = (const float*)d_in[0];      // [50000,128] f32
    const float*     kernel = (const float*)d_in[1];      // [3,128,128] f32
    const float*     bias   = (const float*)d_in[2];      // [3,128]     f32
    const float*     ew     = (const float*)d_in[3];      // [3,800000]  f32
    const long long* eidx   = (const long long*)d_in[4];  // [3,2,800000] i64
    float* out = (float*)d_out;                           // [50000,128] f32
    float* h   = (float*)d_ws;                            // 3*50000*128 f32 = 76.8 MB

    const int nOut = N_NODES * UNITS;
    cheb_init_out<<<(nOut + 255) / 256, 256, 0, stream>>>(out, bias);

    const long tiles = (long)(N_NODES / 16) * (UNITS / 16) * N_HOPS;   // 75000
    cheb_gemm<<<(unsigned)((tiles + 7) / 8), 256, 0, stream>>>(x, kernel, h);

    const long long waves = (long long)N_HOPS * N_EDGES;               // 2.4M
    cheb_scatter<<<(unsigned)((waves + 7) / 8), 256, 0, stream>>>(h, ew, eidx, out);

    cheb_relu<<<(nOut + 255) / 256, 256, 0, stream>>>(out);
}